// Entropy_bottleneck_34136400068724
// MI455X (gfx1250) — compile-verified
//
#include <hip/hip_runtime.h>
#include <cstdint>

#define CH      192
#define BATCH   8
#define HW      4096                  // 64*64
#define NSLAB   (BATCH * CH)          // 1536 blocks
#define TOT     (NSLAB * HW)          // 6291456 elements
#define TPB     256
#define CHUNK_ELTS (TPB * 4)          // 1024 floats per chunk (16B per lane)
#define NCHUNK  (HW / CHUNK_ELTS)     // 4
#define LIKELIHOOD_BOUND 1e-6f

// ---------- math helpers ----------
__device__ __forceinline__ float fast_tanh(float x) {
#if __has_builtin(__builtin_amdgcn_tanhf)
    return __builtin_amdgcn_tanhf(x);      // v_tanh_f32 on gfx1250
#else
    return tanhf(x);
#endif
}

__device__ __forceinline__ float sigmoidf(float x) {
    return 1.0f / (1.0f + __expf(-x));
}

__device__ __forceinline__ float softplusf(float x) {
    // stable softplus: max(x,0) + log(1 + exp(-|x|))
    // log argument is in [1,2] -> no cancellation; __expf/__logf lower to
    // v_exp_f32/v_log_f32 (or v_s_* scalar forms for uniform params) instead
    // of the multi-hundred-instruction libm log1pf expansion.
    return fmaxf(x, 0.0f) + __logf(1.0f + __expf(-fabsf(x)));
}

// ---------- CDNA5 async global->LDS helpers (inline asm; CDNA5 ISA) ----------
__device__ __forceinline__ void async_load_b128(uint32_t lds_off, const void* gptr) {
    // GV mode: VDST = per-lane LDS byte address, VADDR = 64-bit global address
    asm volatile("global_load_async_to_lds_b128 %0, %1, off"
                 :: "v"(lds_off), "v"((uint64_t)(uintptr_t)gptr)
                 : "memory");
}
__device__ __forceinline__ void wait_async1() {
    asm volatile("s_wait_asynccnt 1" ::: "memory");
}
__device__ __forceinline__ void wait_async0() {
    asm volatile("s_wait_asynccnt 0" ::: "memory");
}

// ---------- per-channel transformed parameters (uniform across block) ----------
struct Params {
    float sm0[3], b0[3], t0[3];
    float sm1[9], b1[3], t1[3];
    float sm2[9], b2[3], t2[3];
    float sm3[3], b3;
};

__device__ __forceinline__ float logits_chain(const Params& P, float u) {
    float v[3], w[3];
#pragma unroll
    for (int j = 0; j < 3; ++j) {
        float t = fmaf(P.sm0[j], u, P.b0[j]);
        v[j] = fmaf(P.t0[j], fast_tanh(t), t);
    }
#pragma unroll
    for (int j = 0; j < 3; ++j) {
        float t = P.b1[j];
        t = fmaf(P.sm1[j * 3 + 0], v[0], t);
        t = fmaf(P.sm1[j * 3 + 1], v[1], t);
        t = fmaf(P.sm1[j * 3 + 2], v[2], t);
        w[j] = fmaf(P.t1[j], fast_tanh(t), t);
    }
#pragma unroll
    for (int j = 0; j < 3; ++j) {
        float t = P.b2[j];
        t = fmaf(P.sm2[j * 3 + 0], w[0], t);
        t = fmaf(P.sm2[j * 3 + 1], w[1], t);
        t = fmaf(P.sm2[j * 3 + 2], w[2], t);
        v[j] = fmaf(P.t2[j], fast_tanh(t), t);
    }
    float L = P.b3;
    L = fmaf(P.sm3[0], v[0], L);
    L = fmaf(P.sm3[1], v[1], L);
    L = fmaf(P.sm3[2], v[2], L);
    return L;
}

__global__ __launch_bounds__(TPB) void eb_kernel(
    const float* __restrict__ x,
    const float* __restrict__ m0, const float* __restrict__ m1,
    const float* __restrict__ m2, const float* __restrict__ m3,
    const float* __restrict__ b0, const float* __restrict__ b1,
    const float* __restrict__ b2, const float* __restrict__ b3,
    const float* __restrict__ f0, const float* __restrict__ f1,
    const float* __restrict__ f2,
    const int* __restrict__ training,
    float* __restrict__ out)
{
    __shared__ __align__(16) float stage[2][CHUNK_ELTS];  // 8 KB double buffer

    const int slab = blockIdx.x;          // slab = b*CH + c ; contiguous 4096-float run
    const int c    = slab % CH;
    const int tid  = threadIdx.x;
    const int tr   = training[0];

    // --- load + transform per-channel params once (uniform; scalarized by compiler) ---
    Params P;
    {
        const int c3 = c * 3, c9 = c * 9;
#pragma unroll
        for (int j = 0; j < 3; ++j) {
            P.sm0[j] = softplusf(m0[c3 + j]);
            P.b0[j]  = b0[c3 + j];
            P.t0[j]  = fast_tanh(f0[c3 + j]);
            P.b1[j]  = b1[c3 + j];
            P.t1[j]  = fast_tanh(f1[c3 + j]);
            P.b2[j]  = b2[c3 + j];
            P.t2[j]  = fast_tanh(f2[c3 + j]);
            P.sm3[j] = softplusf(m3[c3 + j]);
        }
#pragma unroll
        for (int j = 0; j < 9; ++j) {
            P.sm1[j] = softplusf(m1[c9 + j]);
            P.sm2[j] = softplusf(m2[c9 + j]);
        }
        P.b3 = b3[c];
    }

    const float* xs = x + (size_t)slab * HW;
    float* xq = out + (size_t)slab * HW;
    float* lk = out + (size_t)TOT + (size_t)slab * HW;

    // low 32 bits of a generic LDS pointer == workgroup-relative LDS byte offset
    const uint32_t lds_base = (uint32_t)(uintptr_t)(&stage[0][0]);
    const uint32_t lane_off = (uint32_t)(tid * 16);

    // prologue: stage chunk 0
    async_load_b128(lds_base + lane_off, xs + tid * 4);

    for (int ck = 0; ck < NCHUNK; ++ck) {
        const int p = ck & 1;
        if (ck + 1 < NCHUNK) {
            // issue next chunk, then wait until the older of the two completes
            async_load_b128(lds_base + (uint32_t)(((ck + 1) & 1) * (CHUNK_ELTS * 4)) + lane_off,
                            xs + (ck + 1) * CHUNK_ELTS + tid * 4);
            wait_async1();
        } else {
            wait_async0();
        }

        const float4 vin = *(const float4*)&stage[p][tid * 4];
        float zi[4] = { vin.x, vin.y, vin.z, vin.w };
        float zo[4], lo[4];

#pragma unroll
        for (int e = 0; e < 4; ++e) {
            float z = zi[e];
            if (tr == 2) z = rintf(z);     // round-half-to-even == jnp.round
            const float lower = logits_chain(P, z - 0.5f);
            const float upper = logits_chain(P, z + 0.5f);
            const float sum = lower + upper;
            const float s = (sum > 0.0f) ? -1.0f : ((sum < 0.0f) ? 1.0f : 0.0f);
            float lik = fabsf(sigmoidf(s * upper) - sigmoidf(s * lower));
            zo[e] = z;
            lo[e] = fmaxf(lik, LIKELIHOOD_BOUND);
        }

        *(float4*)&xq[ck * CHUNK_ELTS + tid * 4] = make_float4(zo[0], zo[1], zo[2], zo[3]);
        *(float4*)&lk[ck * CHUNK_ELTS + tid * 4] = make_float4(lo[0], lo[1], lo[2], lo[3]);
    }
}

extern "C" void kernel_launch(void* const* d_in, const int* in_sizes, int n_in,
                              void* d_out, int out_size, void* d_ws, size_t ws_size,
                              hipStream_t stream) {
    (void)in_sizes; (void)n_in; (void)out_size; (void)d_ws; (void)ws_size;
    const float* x  = (const float*)d_in[0];
    const float* m0 = (const float*)d_in[1];
    const float* m1 = (const float*)d_in[2];
    const float* m2 = (const float*)d_in[3];
    const float* m3 = (const float*)d_in[4];
    const float* b0 = (const float*)d_in[5];
    const float* b1 = (const float*)d_in[6];
    const float* b2 = (const float*)d_in[7];
    const float* b3 = (const float*)d_in[8];
    const float* f0 = (const float*)d_in[9];
    const float* f1 = (const float*)d_in[10];
    const float* f2 = (const float*)d_in[11];
    const int* training = (const int*)d_in[12];
    float* out = (float*)d_out;

    eb_kernel<<<NSLAB, TPB, 0, stream>>>(x, m0, m1, m2, m3,
                                         b0, b1, b2, b3,
                                         f0, f1, f2, training, out);
}